// ProtoLoss_53850299957904
// MI455X (gfx1250) — compile-verified
//
#include <hip/hip_runtime.h>
#include <math.h>

typedef float v2f __attribute__((ext_vector_type(2)));
typedef float v8f __attribute__((ext_vector_type(8)));

#define TEMP_INV 10.0f
#define NEG_BIG  -1e30f
#define SEG_C2   100
#define SEG_C3   20

// Branchless online-logsumexp update (no exec-mask branches in the hot loop).
__device__ __forceinline__ void lse_update(float x, bool valid, bool isPos,
                                           float& m, float& s, float& p)
{
    float cand = valid ? x : NEG_BIG;
    float M    = fmaxf(m, cand);
    float add  = valid ? __expf(cand - M) : 0.f;
    s = s * __expf(m - M) + add;
    m = M;
    p += isPos ? x : 0.f;
}

// ---------------------------------------------------------------------------
// Fine loss: sim = (F @ P1^T)/TEMP fused with row-wise logsumexp + label pick.
// Block = 32 rows, 8 waves. Each wave owns column-tile PAIRS (2w+16i, +1) and
// both 16-row tiles: 4 independent V_WMMA_F32_16X16X4_F32 chains per k-step,
// A reused across 2 col tiles, B reused across 2 row tiles (halves L2 traffic).
// ---------------------------------------------------------------------------
__global__ __launch_bounds__(256)
void fine_loss_kernel(const float* __restrict__ feats,
                      const float* __restrict__ protos,
                      const int*   __restrict__ labels,
                      int C, float* __restrict__ accum)
{
    __shared__ float As[32 * 258];   // 32 rows x 256, stride 258 (bank pad)
    __shared__ int   labs[32];
    __shared__ float wM[8][32];
    __shared__ float wS[8][32];
    __shared__ float wP[8][32];

    const int t       = threadIdx.x;
    const int rowBase = blockIdx.x * 32;

    // Stage 32x256 feature tile into LDS (each thread copies 32 floats)
    {
        const int r  = t >> 3;          // 0..31
        const int c0 = (t & 7) * 32;    // 32 floats per thread
        const float* src = feats + (size_t)(rowBase + r) * 256 + c0;
        float*       dst = &As[r * 258 + c0];
        #pragma unroll
        for (int i = 0; i < 16; ++i) {
            float2 v = ((const float2*)src)[i];
            ((float2*)dst)[i] = v;
        }
    }
    if (t < 32) labs[t] = labels[rowBase + t];
    __syncthreads();

    const int w    = t >> 5;       // wave id 0..7
    const int lane = t & 31;
    const int lrow = lane & 15;    // A row / B col within tile
    const int hi   = lane >> 4;    // K sub-offset selector

    // Per-lane label registers: rows rt*16 + r + 8*hi
    int labR[16];
    #pragma unroll
    for (int rt = 0; rt < 2; ++rt)
        #pragma unroll
        for (int r = 0; r < 8; ++r)
            labR[rt * 8 + r] = labs[rt * 16 + r + 8 * hi];

    // Online logsumexp state for 16 rows per lane
    float m[16], s[16], p[16];
    #pragma unroll
    for (int i = 0; i < 16; ++i) { m[i] = NEG_BIG; s[i] = 0.f; p[i] = 0.f; }

    const float* aBase0 = &As[lrow * 258 + 2 * hi];
    const float* aBase1 = &As[(16 + lrow) * 258 + 2 * hi];
    const int    nTiles = (C + 15) >> 4;   // 63 for C=1000

    for (int ct0 = 2 * w; ct0 < nTiles; ct0 += 16) {
        const int  col0 = ct0 * 16 + lrow;
        const int  col1 = col0 + 16;                  // tile ct0+1 (may be phantom)
        const bool v0   = (col0 < C);
        const bool v1   = (col1 < C);
        const int  bc0  = v0 ? col0 : (C - 1);
        const int  bc1  = v1 ? col1 : (C - 1);
        const float* bBase0 = protos + (size_t)bc0 * 256 + 2 * hi;
        const float* bBase1 = protos + (size_t)bc1 * 256 + 2 * hi;

        v8f a00 = {0,0,0,0,0,0,0,0}, a01 = {0,0,0,0,0,0,0,0};
        v8f a10 = {0,0,0,0,0,0,0,0}, a11 = {0,0,0,0,0,0,0,0};
        #pragma unroll 8
        for (int k = 0; k < 64; ++k) {
            v2f a0 = *(const v2f*)(aBase0 + k * 4);   // ds_load_b64
            v2f a1 = *(const v2f*)(aBase1 + k * 4);
            v2f b0 = *(const v2f*)(bBase0 + k * 4);   // global_load_b64 (L2)
            v2f b1 = *(const v2f*)(bBase1 + k * 4);
            a00 = __builtin_amdgcn_wmma_f32_16x16x4_f32(false, a0, false, b0,
                                                        (short)0, a00, false, false);
            a01 = __builtin_amdgcn_wmma_f32_16x16x4_f32(false, a0, false, b1,
                                                        (short)0, a01, false, false);
            a10 = __builtin_amdgcn_wmma_f32_16x16x4_f32(false, a1, false, b0,
                                                        (short)0, a10, false, false);
            a11 = __builtin_amdgcn_wmma_f32_16x16x4_f32(false, a1, false, b1,
                                                        (short)0, a11, false, false);
        }

        // Branchless epilogue: C/D layout -> VGPR r holds row (rt*16 + r + 8*hi)
        #pragma unroll
        for (int r = 0; r < 8; ++r) {
            float x;
            x = a00[r] * TEMP_INV;
            lse_update(x, v0, col0 == labR[r],      m[r],      s[r],      p[r]);
            x = a01[r] * TEMP_INV;
            lse_update(x, v1, col1 == labR[r],      m[r],      s[r],      p[r]);
            x = a10[r] * TEMP_INV;
            lse_update(x, v0, col0 == labR[8 + r],  m[8 + r],  s[8 + r],  p[8 + r]);
            x = a11[r] * TEMP_INV;
            lse_update(x, v1, col1 == labR[8 + r],  m[8 + r],  s[8 + r],  p[8 + r]);
        }
    }

    // Reduce across the 16 lanes sharing a row half (wave32; xor<16 stays in half)
    #pragma unroll
    for (int off = 1; off < 16; off <<= 1) {
        #pragma unroll
        for (int i = 0; i < 16; ++i) {
            float om = __shfl_xor(m[i], off);
            float os = __shfl_xor(s[i], off);
            float M  = fmaxf(m[i], om);
            s[i] = s[i] * __expf(m[i] - M) + os * __expf(om - M);
            m[i] = M;
            p[i] += __shfl_xor(p[i], off);
        }
    }

    if (lrow == 0) {
        #pragma unroll
        for (int rt = 0; rt < 2; ++rt)
            #pragma unroll
            for (int r = 0; r < 8; ++r) {
                const int rowm = rt * 16 + r + 8 * hi;
                const int i    = rt * 8 + r;
                wM[w][rowm] = m[i];
                wS[w][rowm] = s[i];
                wP[w][rowm] = p[i];
            }
    }
    __syncthreads();

    // Merge the 8 waves; one thread per row; accumulate (pos - lse)
    if (t < 32) {
        float M = NEG_BIG, S = 0.f, P = 0.f;
        #pragma unroll
        for (int ww = 0; ww < 8; ++ww) {
            float om = wM[ww][t], os = wS[ww][t];
            float nM = fmaxf(M, om);
            S = S * __expf(M - nM) + os * __expf(om - nM);
            M = nM;
            P += wP[ww][t];
        }
        atomicAdd(accum, P - (M + __logf(S)));
    }
}

// ---------------------------------------------------------------------------
// Segment sums for both label sets in one pass over features.
// grid = (D/64 slices, 64 row chunks), LDS fp32 atomic accumulation.
// ---------------------------------------------------------------------------
__global__ __launch_bounds__(256)
void segment_sum_kernel(const float* __restrict__ feats,
                        const int* __restrict__ lab2,
                        const int* __restrict__ lab3,
                        float* __restrict__ sum2,
                        float* __restrict__ sum3,
                        int N)
{
    __shared__ float l2[SEG_C2 * 64];
    __shared__ float l3[SEG_C3 * 64];

    const int d       = threadIdx.x & 63;
    const int rl      = threadIdx.x >> 6;             // 0..3
    const int dimBase = blockIdx.x * 64;
    const int rowsPer = N / gridDim.y;
    const int r0      = blockIdx.y * rowsPer;

    for (int i = threadIdx.x; i < SEG_C2 * 64; i += 256) l2[i] = 0.f;
    for (int i = threadIdx.x; i < SEG_C3 * 64; i += 256) l3[i] = 0.f;
    __syncthreads();

    for (int r = r0 + rl; r < r0 + rowsPer; r += 4) {
        float v = feats[(size_t)r * 256 + dimBase + d];
        atomicAdd(&l2[lab2[r] * 64 + d], v);
        atomicAdd(&l3[lab3[r] * 64 + d], v);
    }
    __syncthreads();

    for (int i = threadIdx.x; i < SEG_C2 * 64; i += 256)
        atomicAdd(&sum2[(i >> 6) * 256 + dimBase + (i & 63)], l2[i]);
    for (int i = threadIdx.x; i < SEG_C3 * 64; i += 256)
        atomicAdd(&sum3[(i >> 6) * 256 + dimBase + (i & 63)], l3[i]);
}

__global__ void count_kernel(const int* __restrict__ lab2,
                             const int* __restrict__ lab3,
                             float* __restrict__ cnt2,
                             float* __restrict__ cnt3, int N)
{
    int i = blockIdx.x * blockDim.x + threadIdx.x;
    if (i < N) {
        atomicAdd(&cnt2[lab2[i]], 1.f);
        atomicAdd(&cnt3[lab3[i]], 1.f);
    }
}

// ---------------------------------------------------------------------------
// proto NCE on class means (tiny: C<=100). One block per row i.
// ---------------------------------------------------------------------------
__global__ __launch_bounds__(128)
void proto_nce_kernel(const float* __restrict__ sums,
                      const float* __restrict__ cnts,
                      const float* __restrict__ protos,
                      int C, float* __restrict__ acc)
{
    __shared__ float meanI[256];
    __shared__ float red[128];
    __shared__ float posv;

    const int i = blockIdx.x, t = threadIdx.x;
    const float inv = 1.0f / cnts[i];
    for (int d = t; d < 256; d += 128)
        meanI[d] = sums[(size_t)i * 256 + d] * inv;
    __syncthreads();

    float simj = NEG_BIG;
    if (t < C) {
        const float* pr = protos + (size_t)t * 256;
        float dot = 0.f;
        for (int d = 0; d < 256; ++d) dot = fmaf(meanI[d], pr[d], dot);
        simj = dot * TEMP_INV;
    }
    if (t == i) posv = simj;
    red[t] = simj;
    __syncthreads();
    for (int st = 64; st > 0; st >>= 1) {
        if (t < st) red[t] = fmaxf(red[t], red[t + st]);
        __syncthreads();
    }
    const float M = red[0];
    __syncthreads();
    red[t] = (t < C) ? __expf(simj - M) : 0.f;
    __syncthreads();
    for (int st = 64; st > 0; st >>= 1) {
        if (t < st) red[t] += red[t + st];
        __syncthreads();
    }
    if (t == 0) atomicAdd(acc, posv - (M + __logf(red[0])));
}

__global__ void finalize_kernel(const float* __restrict__ acc,
                                float* __restrict__ out,
                                int N, int C2, int C3)
{
    if (threadIdx.x == 0 && blockIdx.x == 0) {
        float fine   = -acc[0] / (float)N;
        float coarse = -acc[1] / (float)C2;
        float con    = -acc[2] / (float)C3;
        out[0] = 0.5f * fine + 0.5f * coarse + 0.0f * con;
        out[1] = fine;
        out[2] = coarse;
        out[3] = con;
    }
}

// ---------------------------------------------------------------------------
extern "C" void kernel_launch(void* const* d_in, const int* in_sizes, int n_in,
                              void* d_out, int out_size, void* d_ws, size_t ws_size,
                              hipStream_t stream)
{
    const float* feats = (const float*)d_in[0];
    const float* p1    = (const float*)d_in[1];
    const int*   lab1  = (const int*)  d_in[2];
    const float* p2    = (const float*)d_in[3];
    const int*   lab2  = (const int*)  d_in[4];
    const float* p3    = (const float*)d_in[5];
    const int*   lab3  = (const int*)  d_in[6];

    const int D  = 256;
    const int N  = in_sizes[0] / D;   // 131072
    const int C1 = in_sizes[1] / D;   // 1000
    const int C2 = in_sizes[3] / D;   // 100
    const int C3 = in_sizes[5] / D;   // 20

    float* ws   = (float*)d_ws;
    float* sum2 = ws;                 // C2*D
    float* sum3 = sum2 + C2 * D;      // C3*D
    float* cnt2 = sum3 + C3 * D;      // C2
    float* cnt3 = cnt2 + C2;          // C3
    float* acc  = cnt3 + C3;          // 3: fine, coarse, con

    const size_t zeroBytes = (size_t)(C2 * D + C3 * D + C2 + C3 + 3) * sizeof(float);
    hipMemsetAsync(d_ws, 0, zeroBytes, stream);

    fine_loss_kernel<<<N / 32, 256, 0, stream>>>(feats, p1, lab1, C1, &acc[0]);
    segment_sum_kernel<<<dim3(D / 64, 64), 256, 0, stream>>>(feats, lab2, lab3,
                                                             sum2, sum3, N);
    count_kernel<<<(N + 255) / 256, 256, 0, stream>>>(lab2, lab3, cnt2, cnt3, N);
    proto_nce_kernel<<<C2, 128, 0, stream>>>(sum2, cnt2, p2, C2, &acc[1]);
    proto_nce_kernel<<<C3, 128, 0, stream>>>(sum3, cnt3, p3, C3, &acc[2]);
    finalize_kernel<<<1, 1, 0, stream>>>(acc, (float*)d_out, N, C2, C3);
}